// CategoryAwareGate_6768868458611
// MI455X (gfx1250) — compile-verified
//
#include <hip/hip_runtime.h>

// ---------------------------------------------------------------------------
// CategoryAwareGate fused kernel for MI455X (gfx1250, wave32).
//
// Per pixel: class-softmax(swin), class-softmax(gru); per class a 2->32->2
// MLP gate (fp32 WMMA 16x16x4, bias folded into K), 2-way softmax, fusion.
// One wave handles 32 pixels as two 16-pixel WMMA groups (N dimension =
// pixels, M dimension = hidden units / gate outputs).
// ---------------------------------------------------------------------------

typedef float v2f __attribute__((ext_vector_type(2)));
typedef float v8f __attribute__((ext_vector_type(8)));

#define NCLS   16
#define NHID   32
#define HW     65536            // 256*256
#define NPIX   (8 * HW)         // B*H*W = 524288
#define FUSED_ELEMS (8 * NCLS * HW)   // 8388608
#define PREF_ELEMS  (NCLS * 2)        // 32

// LDS float offsets
#define LDS_W1 0        // [c*64 + u*2 + i]        (C,HID,2)  = 1024 floats
#define LDS_B1 1024     // [c*32 + u]              (C,HID)    =  512 floats
#define LDS_W2 1536     // [c*64 + o*32 + h]       (C,2,HID)  = 1024 floats
#define LDS_B2 2560     // [c*2 + o]               (C,2)      =   32 floats
#define LDS_TOT 2592

__device__ __forceinline__ float shfl32(float x, int src) {
    return __shfl(x, src, 32);
}

__global__ __launch_bounds__(256) void gate_fuse_kernel(
    const float* __restrict__ swin, const float* __restrict__ gru,
    const float* __restrict__ W1, const float* __restrict__ b1,
    const float* __restrict__ W2, const float* __restrict__ b2,
    const float* __restrict__ pref,
    float* __restrict__ fusedOut, float* __restrict__ prefOut,
    float* __restrict__ dwOut)
{
    __shared__ float lds[LDS_TOT];
    for (int i = threadIdx.x; i < 1024; i += 256) lds[LDS_W1 + i] = W1[i];
    for (int i = threadIdx.x; i < 512;  i += 256) lds[LDS_B1 + i] = b1[i];
    for (int i = threadIdx.x; i < 1024; i += 256) lds[LDS_W2 + i] = W2[i];
    if (threadIdx.x < 32) lds[LDS_B2 + threadIdx.x] = b2[threadIdx.x];
    __syncthreads();

    // passthrough output #2 (whole wave 0 of block 0 -> uniform branch, wave32)
    if (blockIdx.x == 0 && threadIdx.x < 32) prefOut[threadIdx.x] = pref[threadIdx.x];

    const int lane = threadIdx.x & 31;
    const int m    = lane & 15;          // row index within 16
    const bool lo  = (lane < 16);

    const int p    = blockIdx.x * 256 + threadIdx.x;   // pixel id (lane-contiguous)
    const int bimg = p >> 16;
    const int hw   = p & (HW - 1);
    const int base = (bimg * NCLS) << 16;              // + c*HW + hw

    // ---- load logits (coalesced, per-class strided) -------------------------
    float sl[NCLS], gl[NCLS], sp[NCLS], gp[NCLS];
    #pragma unroll
    for (int c = 0; c < NCLS; ++c) {
        sl[c] = swin[base + c * HW + hw];
        gl[c] = gru [base + c * HW + hw];
    }
    // ---- class-dim softmax (per lane = per pixel) ---------------------------
    float smax = sl[0], gmax = gl[0];
    #pragma unroll
    for (int c = 1; c < NCLS; ++c) { smax = fmaxf(smax, sl[c]); gmax = fmaxf(gmax, gl[c]); }
    float ssum = 0.f, gsum = 0.f;
    #pragma unroll
    for (int c = 0; c < NCLS; ++c) {
        sp[c] = __expf(sl[c] - smax); ssum += sp[c];
        gp[c] = __expf(gl[c] - gmax); gsum += gp[c];
    }
    const float sinv = 1.0f / ssum, ginv = 1.0f / gsum;
    #pragma unroll
    for (int c = 0; c < NCLS; ++c) { sp[c] *= sinv; gp[c] *= ginv; }

    // ---- per-class gate MLP via WMMA ---------------------------------------
    #pragma unroll 1
    for (int c = 0; c < NCLS; ++c) {
        // Layer-1 A (M=16 hidden units, K=4: [w_s, w_g, bias, 0]), two HID halves.
        // A layout: vgpr0 = K0 (lanes0-15) / K2 (lanes16-31); vgpr1 = K1 / K3.
        v2f A1[2];
        #pragma unroll
        for (int hh = 0; hh < 2; ++hh) {
            const int u  = hh * 16 + m;
            const float wa = lds[LDS_W1 + c * 64 + u * 2 + 0];
            const float wb = lds[LDS_W1 + c * 64 + u * 2 + 1];
            const float bb = lds[LDS_B1 + c * 32 + u];
            A1[hh].x = lo ? wa : bb;     // K0 : K2(bias)
            A1[hh].y = lo ? wb : 0.0f;   // K1 : K3(pad)
        }
        // Layer-2 A slices: rows m (only m<2 are live), K slice 4s + (lo?0:2) + {0,1}
        v2f A2[8];
        {
            const int mc    = (m < 2) ? m : 0;
            const int khalf = lo ? 0 : 2;
            #pragma unroll
            for (int s = 0; s < 8; ++s) {
                const float a0 = lds[LDS_W2 + c * 64 + mc * 32 + 4 * s + khalf + 0];
                const float a1 = lds[LDS_W2 + c * 64 + mc * 32 + 4 * s + khalf + 1];
                A2[s].x = (m < 2) ? a0 : 0.0f;
                A2[s].y = (m < 2) ? a1 : 0.0f;
            }
        }
        const float b20 = lds[LDS_B2 + c * 2 + 0];
        const float b21 = lds[LDS_B2 + c * 2 + 1];

        float dwg0 = 0.f, dwg1 = 0.f;
        #pragma unroll
        for (int g = 0; g < 2; ++g) {
            // B1 (K=4 x N=16 pixels of group g): rows [sp, gp, 1, 0]
            // B layout: vgpr0 = row0(lanes0-15)/row1(lanes16-31); vgpr1 = row2/row3.
            v2f B1;
            if (g == 0) {
                const float t = shfl32(gp[c], m);          // gp of pixel (lane&15)
                B1.x = lo ? sp[c] : t;
            } else {
                const float t = shfl32(sp[c], lane | 16);  // sp of pixel (lane|16)
                B1.x = lo ? t : gp[c];
            }
            B1.y = lo ? 1.0f : 0.0f;

            v8f zero = {};
            v8f D0 = __builtin_amdgcn_wmma_f32_16x16x4_f32(false, A1[0], false, B1,
                                                           (short)0, zero, false, false);
            v8f D1 = __builtin_amdgcn_wmma_f32_16x16x4_f32(false, A1[1], false, B1,
                                                           (short)0, zero, false, false);
            #pragma unroll
            for (int r = 0; r < 8; ++r) {           // ReLU
                D0[r] = fmaxf(D0[r], 0.0f);
                D1[r] = fmaxf(D1[r], 0.0f);
            }

            // Layer 2: z = W2 * hid + b2, K=32 as 8 accumulating WMMAs.
            v8f acc = {};
            acc[0] = lo ? b20 : 0.0f;   // M=0 row -> z0 bias
            acc[1] = lo ? b21 : 0.0f;   // M=1 row -> z1 bias
            #pragma unroll
            for (int s = 0; s < 8; ++s) {
                // Build B2 slice (rows 4s..4s+3) from D layout:
                // hid row k lives in D[k>>4] vgpr (k&7), lane-half (k&8).
                v2f B2;
                #pragma unroll
                for (int q = 0; q < 2; ++q) {
                    const int ka = 4 * s + 2 * q;   // goes to lanes 0-15 of this vgpr
                    const int kb = ka + 1;          // goes to lanes 16-31
                    const float da = (ka < 16) ? D0[ka & 7] : D1[ka & 7];
                    const float db = (kb < 16) ? D0[kb & 7] : D1[kb & 7];
                    float v;
                    if ((ka & 8) == 0) {
                        // rows live in low lanes: high lanes fetch row kb from lane-16
                        const float t = shfl32(db, m);
                        v = lo ? da : t;
                    } else {
                        // rows live in high lanes: low lanes fetch row ka from lane+16
                        const float t = shfl32(da, lane | 16);
                        v = lo ? t : db;
                    }
                    if (q == 0) B2.x = v; else B2.y = v;
                }
                acc = __builtin_amdgcn_wmma_f32_16x16x4_f32(false, A2[s], false, B2,
                                                            (short)0, acc, false, false);
            }
            const float z0 = acc[0];   // valid in lanes 0-15 (pixel = 16g + lane)
            const float z1 = acc[1];
            const float w0 = 1.0f / (1.0f + __expf(z1 - z0));   // softmax over 2
            if (g == 0) dwg0 = w0; else dwg1 = w0;
        }

        // Merge groups back to full wave: lanes<16 -> group0, lanes>=16 -> group1.
        const float t   = shfl32(dwg1, m);
        const float dwm = lo ? dwg0 : t;

        fusedOut[base + c * HW + hw] = gl[c] + dwm * (sl[c] - gl[c]);

        float2 pair = make_float2(dwm, 1.0f - dwm);
        *reinterpret_cast<float2*>(dwOut + ((size_t)p * NCLS + c) * 2) = pair;
    }
}

extern "C" void kernel_launch(void* const* d_in, const int* in_sizes, int n_in,
                              void* d_out, int out_size, void* d_ws, size_t ws_size,
                              hipStream_t stream) {
    (void)in_sizes; (void)n_in; (void)out_size; (void)d_ws; (void)ws_size;
    const float* swin = (const float*)d_in[0];
    const float* gru  = (const float*)d_in[1];
    const float* W1   = (const float*)d_in[2];
    const float* b1   = (const float*)d_in[3];
    const float* W2   = (const float*)d_in[4];
    const float* b2   = (const float*)d_in[5];
    const float* pref = (const float*)d_in[6];

    float* out      = (float*)d_out;
    float* fusedOut = out;                              // 8*16*256*256
    float* prefOut  = out + FUSED_ELEMS;                // 32
    float* dwOut    = out + FUSED_ELEMS + PREF_ELEMS;   // N*16*2

    dim3 grid(NPIX / 256), block(256);
    gate_fuse_kernel<<<grid, block, 0, stream>>>(swin, gru, W1, b1, W2, b2, pref,
                                                 fusedOut, prefOut, dwOut);
}